// signedconvolutioninit_6871947673678
// MI455X (gfx1250) — compile-verified
//
#include <hip/hip_runtime.h>
#include <math.h>

typedef float v2f __attribute__((ext_vector_type(2)));
typedef float v8f __attribute__((ext_vector_type(8)));

#define DFEAT 128
#define DOUT  256
#define ROWS  64          // rows (nodes) per block in the GEMM kernel
#define APAD  260         // LDS row stride in dwords; 260 mod 64 == 4 -> conflict-free

// ---------------------------------------------------------------- zero ws
__global__ void zero_ws_kernel(float4* __restrict__ ws, int n4) {
  int i = blockIdx.x * blockDim.x + threadIdx.x;
  if (i < n4) ws[i] = make_float4(0.f, 0.f, 0.f, 0.f);
}

// ------------------------------------------------------- edge scatter-sum
// One wave32 per edge: lane l handles 4 consecutive floats of the 128-wide
// neighbor row (coalesced 512B read), atomicAdd into sums[row] (L2-resident).
// Self-loops (row==col) contribute 0 to sums and counts -> skip.
__global__ void edge_scatter_kernel(const float* __restrict__ feat,
                                    const long long* __restrict__ ei,
                                    float* __restrict__ sums,
                                    float* __restrict__ counts,
                                    int E) {
  int gwave = (blockIdx.x * blockDim.x + threadIdx.x) >> 5;
  int lane  = threadIdx.x & 31;
  if (gwave >= E) return;
  long long row = ei[gwave];
  long long col = ei[(long long)E + gwave];
  if (row == col) return;
  const float4* src = (const float4*)(feat + col * DFEAT);
  float4 v = src[lane];
  float* dst = sums + row * DFEAT + lane * 4;
  atomicAdd(dst + 0, v.x);
  atomicAdd(dst + 1, v.y);
  atomicAdd(dst + 2, v.z);
  atomicAdd(dst + 3, v.w);
  if (lane == 0) atomicAdd(&counts[row], 1.0f);
}

// --------------------------------------- finalize + GEMM (WMMA) + L2 norm
// Block = 256 threads (8 waves) handles ROWS=64 nodes x 256 output cols.
// Wave w owns column tiles [w*32, w*32+16) and [w*32+16, w*32+32) across all
// 4 row tiles -> 8 independent accumulators -> 8 independent v_wmma per
// k-step (no RAW chain through the XDL pipe). A-frags shared across both
// column tiles; B-frags shared across 4 row tiles.
__global__ void __launch_bounds__(256)
gemm_norm_kernel(const float* __restrict__ feat,
                 const float* __restrict__ sums,
                 const float* __restrict__ counts,
                 const float* __restrict__ W,
                 const float* __restrict__ bias,
                 float* __restrict__ out,
                 int N) {
  __shared__ float Atile[ROWS][APAD];   // A tile; reused as result tile after GEMM
  __shared__ float red[ROWS][4];
  __shared__ float cscale[ROWS];
  __shared__ float invn[ROWS];

  const int tid  = threadIdx.x;
  const int row0 = blockIdx.x * ROWS;

  if (tid < ROWS) {
    int node = row0 + tid;
    float c = (node < N) ? counts[node] : 1.0f;
    cscale[tid] = 1.0f / fmaxf(c, 1.0f);
  }
  __syncthreads();

  // Fill A tile: col = tid (0..255), loop over 64 rows. Coalesced loads.
  {
    const int col = tid;
    for (int r = 0; r < ROWS; ++r) {
      int node = row0 + r;
      float v = 0.0f;
      if (node < N) {
        if (col < DFEAT) v = sums[(long long)node * DFEAT + col] * cscale[r];
        else             v = feat[(long long)node * DFEAT + (col - DFEAT)];
      }
      Atile[r][col] = v;
    }
  }
  __syncthreads();

  const int wave = tid >> 5;
  const int lane = tid & 31;
  const int half = lane >> 4;    // 0: K pair (0,1); 1: K pair (2,3)
  const int r16  = lane & 15;    // A row-in-tile / B,D column-in-tile
  const int n0   = wave * 32;    // this wave's 32-column stripe

  v8f acc[4][2];
#pragma unroll
  for (int rt = 0; rt < 4; ++rt)
#pragma unroll
    for (int t = 0; t < 2; ++t)
      acc[rt][t] = (v8f){0.f, 0.f, 0.f, 0.f, 0.f, 0.f, 0.f, 0.f};

  for (int k0 = 0; k0 < DOUT; k0 += 4) {
    const int kk = k0 + 2 * half;
    v2f b0, b1;
    b0.x = W[(long long)kk * DOUT + n0 + r16];
    b0.y = W[(long long)(kk + 1) * DOUT + n0 + r16];
    b1.x = W[(long long)kk * DOUT + n0 + 16 + r16];
    b1.y = W[(long long)(kk + 1) * DOUT + n0 + 16 + r16];
#pragma unroll
    for (int rt = 0; rt < 4; ++rt) {
      v2f a;
      a.x = Atile[rt * 16 + r16][kk];
      a.y = Atile[rt * 16 + r16][kk + 1];
      acc[rt][0] = __builtin_amdgcn_wmma_f32_16x16x4_f32(
          false, a, false, b0, (short)0, acc[rt][0], false, false);
      acc[rt][1] = __builtin_amdgcn_wmma_f32_16x16x4_f32(
          false, a, false, b1, (short)0, acc[rt][1], false, false);
    }
  }
  __syncthreads();   // all A reads done -> reuse Atile as result tile

  {
    const float bb0 = bias[n0 + r16];
    const float bb1 = bias[n0 + 16 + r16];
#pragma unroll
    for (int rt = 0; rt < 4; ++rt)
#pragma unroll
      for (int j = 0; j < 8; ++j) {
        // VGPR j, lane half -> row (j + 8*half) of tile rt, col n0(+16) + r16
        Atile[rt * 16 + j + 8 * half][n0 + r16]      = acc[rt][0][j] + bb0;
        Atile[rt * 16 + j + 8 * half][n0 + 16 + r16] = acc[rt][1][j] + bb1;
      }
  }
  __syncthreads();

  // Row-wise L2 norm: 4 threads per row; stride-4 column access keeps all 32
  // lanes of a wave on distinct LDS banks (bank = 4*row + seg + 4i mod 64).
  const int rowi = tid >> 2;
  const int seg  = tid & 3;
  float s = 0.f;
  for (int i = 0; i < DOUT / 4; ++i) {
    float v = Atile[rowi][seg + 4 * i];
    s += v * v;
  }
  red[rowi][seg] = s;
  __syncthreads();
  if (tid < ROWS) {
    float tot = red[tid][0] + red[tid][1] + red[tid][2] + red[tid][3];
    invn[tid] = 1.0f / fmaxf(sqrtf(tot), 1e-12f);
  }
  __syncthreads();

  // Store: thread writes 64 contiguous floats of its row (16 float4),
  // 4 threads cover a full 256-wide row -> fully coalesced.
  if (row0 + rowi < N) {
    const float iv = invn[rowi];
    float4* op = (float4*)(out + (long long)(row0 + rowi) * DOUT + seg * 64);
#pragma unroll
    for (int q = 0; q < 16; ++q) {
      float4 o;
      o.x = Atile[rowi][seg * 64 + q * 4 + 0] * iv;
      o.y = Atile[rowi][seg * 64 + q * 4 + 1] * iv;
      o.z = Atile[rowi][seg * 64 + q * 4 + 2] * iv;
      o.w = Atile[rowi][seg * 64 + q * 4 + 3] * iv;
      op[q] = o;
    }
  }
}

// ---------------------------------------------------------------- launch
extern "C" void kernel_launch(void* const* d_in, const int* in_sizes, int n_in,
                              void* d_out, int out_size, void* d_ws, size_t ws_size,
                              hipStream_t stream) {
  const float*     feat = (const float*)d_in[0];
  const long long* ei   = (const long long*)d_in[1];
  const float*     W    = (const float*)d_in[2];
  const float*     bias = (const float*)d_in[3];
  float*           out  = (float*)d_out;

  const int N = in_sizes[0] / DFEAT;   // 100000
  const int E = in_sizes[1] / 2;       // 600000

  float* sums   = (float*)d_ws;                 // N*128 floats
  float* counts = sums + (size_t)N * DFEAT;     // N floats

  // 1) zero sums + counts
  {
    long long nfl = (long long)N * DFEAT + N;
    int n4 = (int)((nfl + 3) / 4);
    int grid = (n4 + 255) / 256;
    zero_ws_kernel<<<grid, 256, 0, stream>>>((float4*)d_ws, n4);
  }

  // 2) edge scatter (one wave per edge, 8 edges per 256-thread block)
  {
    int grid = (E + 7) / 8;
    edge_scatter_kernel<<<grid, 256, 0, stream>>>(feat, ei, sums, counts, E);
  }

  // 3) fused finalize + GEMM + bias + L2 normalize (64 rows per block)
  {
    int grid = (N + ROWS - 1) / ROWS;   // 1563
    gemm_norm_kernel<<<grid, 256, 0, stream>>>(feat, sums, counts, W, bias, out, N);
  }
}